// RouteSelectModel_89928025244066
// MI455X (gfx1250) — compile-verified
//
#include <hip/hip_runtime.h>
#include <hip/hip_bf16.h>

// ---------------------------------------------------------------------------
// GAT + deep MLP for gfx1250 (CDNA5).
// GEMMs: V_WMMA_F32_16X16X4_F32 (full f32 precision, matches f32 reference),
// with weights pre-packed into zero-padded, K-pair-interleaved buffers so the
// inner loop is branch-free b64 loads + WMMAs only.
// Edge softmax: 3-pass segmented softmax on L2 atomics
// (global_atomic_max_u32 on order-preserving keys + global_atomic_add_f32).
// ---------------------------------------------------------------------------

typedef float v2f __attribute__((ext_vector_type(2)));
typedef float v8f __attribute__((ext_vector_type(8)));

// Order-preserving float -> uint key so segment-max lowers to one atomic_max_u32.
__device__ __forceinline__ unsigned fkey(float f) {
  unsigned u = __float_as_uint(f);
  return (u >> 31) ? ~u : (u | 0x80000000u);
}
__device__ __forceinline__ float fkey_inv(unsigned k) {
  unsigned u = (k >> 31) ? (k ^ 0x80000000u) : ~k;
  return __uint_as_float(u);
}
#define FKEY_NEG_INF 0x007FFFFFu   // fkey(-inf)

// ---------------------------------------------------------------------------
// Weight packing: W[K,Nact] row-major -> Bp, zero-padded to [Kp, Npp] and
// pair-interleaved along K:  Bp[(k>>1)*2*Npp + n*2 + (k&1)] = W[k][n].
// This makes each lane's (B[ka][col], B[ka+1][col]) fragment one b64 load.
// ---------------------------------------------------------------------------
__global__ void pack_weights(const float* __restrict__ W, float* __restrict__ Bp,
                             int K, int Nact, int Kp, int Npp)
{
  const int i = blockIdx.x * blockDim.x + threadIdx.x;
  if (i >= Kp * Npp) return;
  const int k = i / Npp;
  const int n = i - k * Npp;
  const float v = (k < K && n < Nact) ? W[k * Nact + n] : 0.0f;
  Bp[(k >> 1) * (2 * Npp) + n * 2 + (k & 1)] = v;
}

// ---------------------------------------------------------------------------
// f32 WMMA GEMM: C[M, Nact] = op(A[M,Kp] * B + bias). One wave per 16x(NT*16)
// output strip (A fragment reused across NT tiles). All hot-loop loads are
// unconditional global_load_b64:
//   A-frag (16x4 f32): lane l holds A[tm*16 + (l&15)][k0 + 2*(l>>4) + {0,1}]
//   B-frag (4x16 f32): lane l holds B[k0 + 2*(l>>4) + {0,1}][coltile + (l&15)]
//   C/D (16x16 f32):   vgpr j of lane l = C[tm*16 + j + 8*(l>>4)][...]
// Requirements guaranteed by the launcher:
//   Kp % 4 == 0, Kp <= lda, A pad columns are zero & finite,
//   Npp >= gridDim.y * NT * 16 (packed B covers every loaded column).
// ---------------------------------------------------------------------------
template <int NT>
__global__ void wmma_gemm_f32(const float* __restrict__ A, int lda,
                              const float* __restrict__ Bp, int Npp,
                              const float* __restrict__ bias,
                              float* __restrict__ C, int ldc,
                              int Kp, int Nact, int do_relu)
{
  const int lane = threadIdx.x & 31;
  const int half = lane >> 4;       // 0: K lanes {0,1}; 1: K lanes {2,3}
  const int l16  = lane & 15;
  const long tm  = blockIdx.x;
  const int tn0  = blockIdx.y * NT;

  const float* Ap = A + (tm * 16 + l16) * (long)lda + 2 * half;
  const float* Bq = Bp + half * (2 * Npp) + (tn0 * 16 + l16) * 2;

  v8f zero = {};
  v8f acc[NT];
#pragma unroll
  for (int t = 0; t < NT; ++t) acc[t] = zero;

  for (int k0 = 0; k0 < Kp; k0 += 4) {
    const float2 av = *(const float2*)Ap;
    v2f a;
    a.x = av.x;
    a.y = av.y;
#pragma unroll
    for (int t = 0; t < NT; ++t) {
      const float2 bv = *(const float2*)(Bq + t * 32);
      v2f b;
      b.x = bv.x;
      b.y = bv.y;
      acc[t] = __builtin_amdgcn_wmma_f32_16x16x4_f32(
          /*neg_a=*/false, a, /*neg_b=*/false, b,
          /*c_mod=*/(short)0, acc[t], /*reuse_a=*/false, /*reuse_b=*/false);
    }
    Ap += 4;
    Bq += 4 * (long)Npp;
  }

  const long row0 = tm * 16;
#pragma unroll
  for (int t = 0; t < NT; ++t) {
    if ((tn0 + t) * 16 < ldc) {                 // uniform per-tile guard
      const int colt = (tn0 + t) * 16 + l16;
      const bool cok = colt < Nact;
      const float bv = bias ? bias[cok ? colt : (Nact - 1)] : 0.0f;
#pragma unroll
      for (int j = 0; j < 8; ++j) {
        float v = acc[t][j] + bv;
        if (do_relu) v = fmaxf(v, 0.0f);
        v = cok ? v : 0.0f;                      // zero the pad columns
        C[(row0 + j + 8 * half) * (long)ldc + colt] = v;
      }
    }
  }
}

// el/er: per (node, head) dot of feat[n,h,:] with attention vectors.
__global__ void attn_scores(const float* __restrict__ feat,
                            const float* __restrict__ attn_l,
                            const float* __restrict__ attn_r,
                            float* __restrict__ el, float* __restrict__ er, int NH)
{
  const int i = blockIdx.x * blockDim.x + threadIdx.x;  // n*4 + h
  if (i >= NH) return;
  const int h = i & 3;
  const float* fp = feat + (long)(i >> 2) * 64 + h * 16;
  float sl = 0.0f, sr = 0.0f;
#pragma unroll
  for (int f = 0; f < 16; ++f) {
    const float v = fp[f];
    sl += v * attn_l[h * 16 + f];
    sr += v * attn_r[h * 16 + f];
  }
  el[i] = sl;
  er[i] = sr;
}

__global__ void init_bufs(unsigned* __restrict__ mkey, float* __restrict__ den,
                          float* __restrict__ out, int NH, long N64)
{
  const long i = (long)blockIdx.x * blockDim.x + threadIdx.x;
  if (i < NH) { mkey[i] = FKEY_NEG_INF; den[i] = 0.0f; }
  if (i < N64) out[i] = 0.0f;
}

// Pass 1: e = leaky_relu(el[src]+er[dst]); segment max over dst.
__global__ void edge_lrelu_max(const int* __restrict__ src, const int* __restrict__ dst,
                               const float* __restrict__ el, const float* __restrict__ er,
                               float* __restrict__ eW, unsigned* __restrict__ mkey, long EH)
{
  const long i = (long)blockIdx.x * blockDim.x + threadIdx.x;
  if (i >= EH) return;
  const long e = i >> 2;
  const int  h = (int)(i & 3);
  const int  s = src[e], d = dst[e];
  float v = el[s * 4 + h] + er[d * 4 + h];
  v = v > 0.0f ? v : 0.2f * v;
  eW[i] = v;
  atomicMax(&mkey[d * 4 + h], fkey(v));
}

// Pass 2: ez = exp(e - m[dst]); segment sum.
__global__ void edge_exp_sum(const int* __restrict__ dst,
                             const unsigned* __restrict__ mkey,
                             float* __restrict__ eW, float* __restrict__ den, long EH)
{
  const long i = (long)blockIdx.x * blockDim.x + threadIdx.x;
  if (i >= EH) return;
  const long e = i >> 2;
  const int  h = (int)(i & 3);
  const int  d = dst[e];
  const float mv = fkey_inv(mkey[d * 4 + h]);
  const float ez = __expf(eW[i] - mv);
  eW[i] = ez;
  atomicAdd(&den[d * 4 + h], ez);
}

// Pass 3: out[dst] += (ez/denom[dst]) * feat[src]. One wave per edge; each
// lane owns one float2 of the 64-wide feature vector (pairs stay in-head).
__global__ void edge_aggregate(const int* __restrict__ src, const int* __restrict__ dst,
                               const float* __restrict__ feat,
                               const float* __restrict__ eW, const float* __restrict__ den,
                               float* __restrict__ out, long E)
{
  const long tid  = (long)blockIdx.x * blockDim.x + threadIdx.x;
  const long e    = tid >> 5;
  const int  lane = threadIdx.x & 31;
  if (e >= E) return;
  const int s = src[e], d = dst[e];
  const int c = lane * 2;
  const int h = c >> 4;
  const float alpha = eW[e * 4 + h] / den[d * 4 + h];
  const float2 f = *(const float2*)(feat + (long)s * 64 + c);
  atomicAdd(out + (long)d * 64 + c,     alpha * f.x);
  atomicAdd(out + (long)d * 64 + c + 1, alpha * f.y);
}

__global__ void add_bias64(float* __restrict__ out, const float* __restrict__ bias, long N64)
{
  const long i = (long)blockIdx.x * blockDim.x + threadIdx.x;
  if (i < N64) out[i] += bias[i & 63];
}

// Final: h7 = a6 @ w7 + b7 (32->1)  plus  wx = g @ wt (640->1).
__global__ void mlp_final(const float* __restrict__ a6, const float* __restrict__ w7,
                          const float* __restrict__ b7, const float* __restrict__ g,
                          const float* __restrict__ wt, float* __restrict__ out, int P)
{
  const int p = blockIdx.x * blockDim.x + threadIdx.x;
  if (p >= P) return;
  float acc = b7[0];
#pragma unroll
  for (int k = 0; k < 32; ++k) acc += a6[(long)p * 32 + k] * w7[k];
  for (int k = 0; k < 640; ++k) acc += g[(long)p * 640 + k] * wt[k];
  out[p] = acc;
}

// ---------------------------------------------------------------------------
extern "C" void kernel_launch(void* const* d_in, const int* in_sizes, int n_in,
                              void* d_out, int out_size, void* d_ws, size_t ws_size,
                              hipStream_t stream)
{
  const float* x      = (const float*)d_in[0];
  const int*   src    = (const int*)  d_in[1];
  const int*   dst    = (const int*)  d_in[2];
  const float* fc_w   = (const float*)d_in[3];
  const float* bias   = (const float*)d_in[4];
  const float* attn_l = (const float*)d_in[5];
  const float* attn_r = (const float*)d_in[6];
  const float* W[7];
  const float* Bv[7];
  for (int i = 0; i < 7; ++i) { W[i] = (const float*)d_in[7 + 2 * i]; Bv[i] = (const float*)d_in[8 + 2 * i]; }
  const float* wt = (const float*)d_in[21];

  const int  N   = in_sizes[0] / 64;   // 100000
  const long E   = in_sizes[1];        // 1700000
  const int  P   = N / 10;             // 10000 paths
  const long N64 = (long)N * 64;
  const long EH  = E * 4;
  const int  T   = 256;

  // ---- workspace layout (floats) ----
  float* ws = (float*)d_ws;
  size_t o = 0;
  float*    feat = ws + o;              o += (size_t)N * 64;
  float*    el   = ws + o;              o += (size_t)N * 4;
  float*    er   = ws + o;              o += (size_t)N * 4;
  unsigned* mkey = (unsigned*)(ws + o); o += (size_t)N * 4;
  float*    den  = ws + o;              o += (size_t)N * 4;
  float*    g    = ws + o;              o += (size_t)N * 64;  // GAT out == MLP in

  // packed weights: {K, Nact, Kp, Npp} per GEMM (fc + 6 MLP layers)
  //   fc : 64 ->64   Kp 64  Npp 64
  //   L1 : 640->180  Kp 640 Npp 192
  //   L2 : 180->150  Kp 180 Npp 192
  //   L3 : 150->128  Kp 152 Npp 128
  //   L4 : 128->80   Kp 128 Npp 128
  //   L5 : 80 ->64   Kp 80  Npp 64
  //   L6 : 64 ->32   Kp 64  Npp 32
  const int Kact[7] = { 64, 640, 180, 150, 128, 80, 64 };
  const int Nact[7] = { 64, 180, 150, 128,  80, 64, 32 };
  const int Kp[7]   = { 64, 640, 180, 152, 128, 80, 64 };
  const int Npp[7]  = { 64, 192, 192, 128, 128, 64, 32 };
  float* pw[7];
  for (int i = 0; i < 7; ++i) { pw[i] = ws + o; o += (size_t)Kp[i] * Npp[i]; }

  float* eW = ws + o;                   // E*4 floats; MLP activations overlay
  float* a1 = eW;                       // [P,192] (180 valid, pad zeroed)
  float* a2 = a1 + (size_t)P * 192;     // [P,160] (150 valid, pad zeroed)
  float* a3 = a2 + (size_t)P * 160;     // [P,128]
  float* a4 = a3 + (size_t)P * 128;     // [P, 80]
  float* a5 = a4 + (size_t)P * 80;      // [P, 64]
  float* a6 = a5 + (size_t)P * 64;      // [P, 32]

  // 0) pack + zero-pad all weight matrices (pair-interleaved along K)
  pack_weights<<<(64 * 64 + T - 1) / T, T, 0, stream>>>(fc_w, pw[0], 64, 64, 64, 64);
  for (int i = 1; i < 7; ++i) {
    const int elems = Kp[i] * Npp[i];
    pack_weights<<<(elems + T - 1) / T, T, 0, stream>>>(W[i - 1], pw[i],
                                                        Kact[i], Nact[i], Kp[i], Npp[i]);
  }

  // 1) feat = x @ fc_w   [N,64]  (f32 WMMA, NT=4 -> grid.y = 1)
  wmma_gemm_f32<4><<<dim3(N / 16, 1), 32, 0, stream>>>(x, 64, pw[0], 64, nullptr,
                                                       feat, 64, 64, 64, 0);
  // 2) attention scores
  attn_scores<<<(N * 4 + T - 1) / T, T, 0, stream>>>(feat, attn_l, attn_r, el, er, N * 4);
  // 3) init segment buffers + output accumulator
  init_bufs<<<(unsigned)((N64 + T - 1) / T), T, 0, stream>>>(mkey, den, g, N * 4, N64);
  // 4) edge leaky-relu + segment max
  edge_lrelu_max<<<(unsigned)((EH + T - 1) / T), T, 0, stream>>>(src, dst, el, er, eW, mkey, EH);
  // 5) exp + segment sum
  edge_exp_sum<<<(unsigned)((EH + T - 1) / T), T, 0, stream>>>(dst, mkey, eW, den, EH);
  // 6) weighted scatter-add aggregation (one wave per edge)
  edge_aggregate<<<(unsigned)((E * 32 + T - 1) / T), T, 0, stream>>>(src, dst, feat, eW, den, g, E);
  // 7) post-aggregation bias
  add_bias64<<<(unsigned)((N64 + T - 1) / T), T, 0, stream>>>(g, bias, N64);

  // 8-13) deep MLP (all f32 WMMA; branch-free inner loops)
  const int PT = P / 16;  // 625 row tiles
  wmma_gemm_f32<4><<<dim3(PT, 3), 32, 0, stream>>>(g, 640, pw[1], 192, Bv[0], a1, 192, 640, 180, 1);
  wmma_gemm_f32<4><<<dim3(PT, 3), 32, 0, stream>>>(a1, 192, pw[2], 192, Bv[1], a2, 160, 180, 150, 1);
  wmma_gemm_f32<4><<<dim3(PT, 2), 32, 0, stream>>>(a2, 160, pw[3], 128, Bv[2], a3, 128, 152, 128, 1);
  wmma_gemm_f32<4><<<dim3(PT, 2), 32, 0, stream>>>(a3, 128, pw[4], 128, Bv[3], a4,  80, 128,  80, 1);
  wmma_gemm_f32<4><<<dim3(PT, 1), 32, 0, stream>>>(a4,  80, pw[5],  64, Bv[4], a5,  64,  80,  64, 1);
  wmma_gemm_f32<2><<<dim3(PT, 1), 32, 0, stream>>>(a5,  64, pw[6],  32, Bv[5], a6,  32,  64,  32, 1);

  // 14) last layer (32->1) + linear_trans (640->1)
  mlp_final<<<(P + T - 1) / T, T, 0, stream>>>(a6, W[6], Bv[6], g, wt, (float*)d_out, P);
}